// BatchQuantumLayer_43112881717902
// MI455X (gfx1250) — compile-verified
//
#include <hip/hip_runtime.h>
#include <math.h>

#define N_QUBITS 14
#define DIM 16384          // 2^14
#define BATCH 1024
#define N_LAYERS 6
#define PI_F 3.14159265358979f

typedef float v2f __attribute__((ext_vector_type(2)));
typedef float v8f __attribute__((ext_vector_type(8)));

// ---------------------------------------------------------------------------
// Kernel 1: per-feature min/max over the batch -> ws[0..13]=min, ws[14..27]=max
// ---------------------------------------------------------------------------
__global__ void __launch_bounds__(256)
minmax_kernel(const float* __restrict__ x, float* __restrict__ ws)
{
    __shared__ float buf[256];
    const int tid = threadIdx.x;
    float mn[N_QUBITS], mx[N_QUBITS];
#pragma unroll
    for (int f = 0; f < N_QUBITS; ++f) { mn[f] = __builtin_inff(); mx[f] = -__builtin_inff(); }
    for (int r = tid; r < BATCH; r += 256) {
#pragma unroll
        for (int f = 0; f < N_QUBITS; ++f) {
            float v = x[r * N_QUBITS + f];
            mn[f] = fminf(mn[f], v);
            mx[f] = fmaxf(mx[f], v);
        }
    }
    for (int f = 0; f < N_QUBITS; ++f) {
        buf[tid] = mn[f]; __syncthreads();
        for (int s = 128; s > 0; s >>= 1) { if (tid < s) buf[tid] = fminf(buf[tid], buf[tid + s]); __syncthreads(); }
        if (tid == 0) ws[f] = buf[0];
        __syncthreads();
        buf[tid] = mx[f]; __syncthreads();
        for (int s = 128; s > 0; s >>= 1) { if (tid < s) buf[tid] = fmaxf(buf[tid], buf[tid + s]); __syncthreads(); }
        if (tid == 0) ws[N_QUBITS + f] = buf[0];
        __syncthreads();
    }
}

// ---------------------------------------------------------------------------
// State index layout: qubit q occupies bit (13-q). A 4-qubit group whose LSB
// sits at bit p occupies bits p+3..p. addr(j,o): j = group index (0..15),
// o = combined other-bits index (0..1023).
// ---------------------------------------------------------------------------
__device__ __forceinline__ int gaddr(int p, int j, int o)
{
    const int lowmask = (1 << p) - 1;
    return ((o & ~lowmask) << 4) | (j << p) | (o & lowmask);
}

// Build 16x16 kron matrix R(t0) (x) R(t1) (x) R(t2) (x) R(t3) into LDS.
// t0 is the most-significant qubit of the group (bit 3 of group index).
__device__ __forceinline__ void build_m16(float* M16, float t0, float t1, float t2, float t3, int tid)
{
    const int i = tid >> 4, j = tid & 15;
    float th[4] = { t0, t1, t2, t3 };
    float prod = 1.0f;
#pragma unroll
    for (int t = 0; t < 4; ++t) {
        float s, c;
        sincosf(0.5f * th[t], &s, &c);
        const int ib = (i >> (3 - t)) & 1;
        const int jb = (j >> (3 - t)) & 1;
        // RY: [[c,-s],[s,c]]
        float r = (ib == 0) ? (jb == 0 ? c : -s) : (jb == 0 ? s : c);
        prod *= r;
    }
    M16[tid] = prod;
}

// Build 4x4 kron matrix R(t12) (x) R(t13) into LDS (bit1<->qubit12, bit0<->qubit13).
__device__ __forceinline__ void build_m4(float* M4, float t12, float t13, int tid)
{
    if (tid < 16) {
        const int i = tid >> 2, j = tid & 3;
        float sA, cA, sB, cB;
        sincosf(0.5f * t12, &sA, &cA);
        sincosf(0.5f * t13, &sB, &cB);
        const int i1 = i >> 1, j1 = j >> 1, i0 = i & 1, j0 = j & 1;
        float rA = (i1 == 0) ? (j1 == 0 ? cA : -sA) : (j1 == 0 ? sA : cA);
        float rB = (i0 == 0) ? (j0 == 0 ? cB : -sB) : (j0 == 0 ? sB : cB);
        M4[tid] = rA * rB;
    }
}

// Apply 16x16 matrix M16 along group axis at bit position p:
//   nxt[addr(i,o)] = sum_j M16[i][j] * cur[addr(j,o)]
// 1024 columns -> 64 tiles of 16; 8 waves x 8 tiles each.
// D(16x16) = A(16x16) x B(16x16) via four chained V_WMMA_F32_16X16X4_F32.
// Operand layouts (wave32):
//   A (16x4): vgpr v, lane l -> M = l%16, K = v + 2*(l/16)
//   B (4x16): vgpr v, lane l -> K = v + 2*(l/16), N = l%16
//   C/D     : vgpr r, lane l -> M = r + 8*(l/16), N = l%16
__device__ __forceinline__ void group_pass(const float* __restrict__ cur,
                                           float* __restrict__ nxt,
                                           const float* __restrict__ M16,
                                           int p, int tid)
{
    const int lane = tid & 31;
    const int wave = tid >> 5;
    const int lh   = lane >> 4;   // lane half (0/1)
    const int lm   = lane & 15;

    v2f a[4];
#pragma unroll
    for (int kc = 0; kc < 4; ++kc) {
        a[kc].x = M16[lm * 16 + (4 * kc + 0 + 2 * lh)];
        a[kc].y = M16[lm * 16 + (4 * kc + 1 + 2 * lh)];
    }

#pragma unroll
    for (int tt = 0; tt < 8; ++tt) {
        const int col = (wave * 8 + tt) * 16 + lm;
        v2f bm[4];
#pragma unroll
        for (int kc = 0; kc < 4; ++kc) {
            bm[kc].x = cur[gaddr(p, 4 * kc + 0 + 2 * lh, col)];
            bm[kc].y = cur[gaddr(p, 4 * kc + 1 + 2 * lh, col)];
        }
        v8f c = { 0.f, 0.f, 0.f, 0.f, 0.f, 0.f, 0.f, 0.f };
#pragma unroll
        for (int kc = 0; kc < 4; ++kc) {
            c = __builtin_amdgcn_wmma_f32_16x16x4_f32(
                /*neg_a=*/false, a[kc], /*neg_b=*/false, bm[kc],
                /*c_mod=*/(short)0, c, /*reuse_a=*/false, /*reuse_b=*/false);
        }
#pragma unroll
        for (int r = 0; r < 8; ++r)
            nxt[gaddr(p, r + 8 * lh, col)] = c[r];
    }
}

// Apply 4x4 matrix on the two least-significant bits (qubits 12,13): quads are contiguous.
__device__ __forceinline__ void p2_pass(const float* __restrict__ cur,
                                        float* __restrict__ nxt,
                                        const float* __restrict__ M4, int tid)
{
    for (int q = tid; q < DIM / 4; q += 256) {
        const int base = q * 4;
        float i0 = cur[base + 0], i1 = cur[base + 1], i2 = cur[base + 2], i3 = cur[base + 3];
#pragma unroll
        for (int i = 0; i < 4; ++i)
            nxt[base + i] = M4[i * 4 + 0] * i0 + M4[i * 4 + 1] * i1 +
                            M4[i * 4 + 2] * i2 + M4[i * 4 + 3] * i3;
    }
}

// ---------------------------------------------------------------------------
// Kernel 2: one block per batch element; full circuit in LDS (double-buffered)
// ---------------------------------------------------------------------------
__global__ void __launch_bounds__(256)
qsim_kernel(const float* __restrict__ x, const float* __restrict__ w,
            const float* __restrict__ ws, float* __restrict__ out)
{
    extern __shared__ float smem[];
    float* S0  = smem;                 // 16384
    float* S1  = smem + DIM;           // 16384
    float* M16 = smem + 2 * DIM;       // 256
    float* M4  = M16 + 256;            // 16
    float* ang = M4 + 16;              // 16

    const int tid = threadIdx.x;
    const int b   = blockIdx.x;

    for (int i = tid; i < DIM; i += 256) S0[i] = 0.0f;
    if (tid == 0) S0[0] = 1.0f;
    if (tid < N_QUBITS) {
        const float mn = ws[tid], mx = ws[N_QUBITS + tid];
        ang[tid] = (x[b * N_QUBITS + tid] - mn) / (mx - mn + 1e-8f) * PI_F;
    }
    __syncthreads();

    float* cur = S0;
    float* nxt = S1;

    // ---- encoding layer: RY(x_norm[q]) on all qubits ----
#pragma unroll
    for (int g = 0; g < 3; ++g) {
        build_m16(M16, ang[g * 4 + 0], ang[g * 4 + 1], ang[g * 4 + 2], ang[g * 4 + 3], tid);
        __syncthreads();
        group_pass(cur, nxt, M16, 10 - 4 * g, tid);
        __syncthreads();
        float* t = cur; cur = nxt; nxt = t;
    }
    build_m4(M4, ang[12], ang[13], tid);
    __syncthreads();
    p2_pass(cur, nxt, M4, tid);
    __syncthreads();
    { float* t = cur; cur = nxt; nxt = t; }

    // ---- variational layers ----
    for (int layer = 0; layer < N_LAYERS; ++layer) {
        const float* wl = w + layer * N_QUBITS;
#pragma unroll
        for (int g = 0; g < 3; ++g) {
            build_m16(M16, wl[g * 4 + 0], wl[g * 4 + 1], wl[g * 4 + 2], wl[g * 4 + 3], tid);
            __syncthreads();
            group_pass(cur, nxt, M16, 10 - 4 * g, tid);
            __syncthreads();
            float* t = cur; cur = nxt; nxt = t;
        }
        build_m4(M4, wl[12], wl[13], tid);
        __syncthreads();
        p2_pass(cur, nxt, M4, tid);
        __syncthreads();
        { float* t = cur; cur = nxt; nxt = t; }

        // CNOT cascade i=0..12 == suffix-XOR permutation of index bits:
        // y_j = XOR_{m>=j} x_m ; amplitude at x moves to y.
        for (int idx = tid; idx < DIM; idx += 256) {
            int y = idx ^ (idx >> 1);
            y ^= y >> 2; y ^= y >> 4; y ^= y >> 8;
            nxt[y] = cur[idx];
        }
        __syncthreads();
        { float* t = cur; cur = nxt; nxt = t; }
    }

    // ---- expectation values <Z_0>, <Z_1> (bits 13, 12) ----
    float e0 = 0.0f, e1 = 0.0f;
    for (int idx = tid; idx < DIM; idx += 256) {
        const float v = cur[idx];
        const float pv = v * v;
        e0 += (idx & (1 << 13)) ? -pv : pv;
        e1 += (idx & (1 << 12)) ? -pv : pv;
    }
    float* red = nxt;  // free buffer as reduction scratch
    red[tid] = e0;
    red[256 + tid] = e1;
    __syncthreads();
    for (int s = 128; s > 0; s >>= 1) {
        if (tid < s) {
            red[tid]       += red[tid + s];
            red[256 + tid] += red[256 + tid + s];
        }
        __syncthreads();
    }
    if (tid == 0) {
        out[b * 2 + 0] = red[0];
        out[b * 2 + 1] = red[256];
    }
}

// ---------------------------------------------------------------------------
extern "C" void kernel_launch(void* const* d_in, const int* in_sizes, int n_in,
                              void* d_out, int out_size, void* d_ws, size_t ws_size,
                              hipStream_t stream)
{
    const float* x  = (const float*)d_in[0];   // (1024, 14)
    const float* w  = (const float*)d_in[1];   // (6, 14)
    float* out = (float*)d_out;                // (1024, 2)
    float* ws  = (float*)d_ws;                 // mn[14], mx[14]

    const size_t shbytes = (size_t)(2 * DIM + 256 + 16 + 16) * sizeof(float);
    // Opt into >64KB dynamic LDS (no-op for graph capture; not a stream op).
    (void)hipFuncSetAttribute((const void*)qsim_kernel,
                              hipFuncAttributeMaxDynamicSharedMemorySize,
                              (int)shbytes);

    minmax_kernel<<<1, 256, 0, stream>>>(x, ws);
    qsim_kernel<<<BATCH, 256, shbytes, stream>>>(x, w, ws, out);
}